// Conv3d_90821378441234
// MI455X (gfx1250) — compile-verified
//
#include <hip/hip_runtime.h>

typedef __attribute__((ext_vector_type(2))) float v2f;
typedef __attribute__((ext_vector_type(8))) float v8f;
typedef int v4i __attribute__((vector_size(16)));   // payload type for async b128

#define SPK     27
#define SPCIN   32
#define SPCOUT  64
#define ROWPAD  36   // padded LDS row stride (floats): 16B-aligned stores, conflict-free b64 reads
#define PACKED_FLOATS (SPK * 4 * 32 * 16)                 // 55296: [k][couttile][lane][16]
#define WS_NEED ((32 + PACKED_FLOATS) * sizeof(float))    // zero row + packed weights

#if defined(__has_builtin)
#  if __has_builtin(__builtin_amdgcn_global_load_async_to_lds_b128)
#    define HAVE_ASYNC_LDS 1
#  endif
#endif

// Issue a 16B global->LDS gather. Async path uses the CDNA5 ASYNCcnt engine;
// fallback goes through VGPRs (load + ds_store).
static __device__ __forceinline__ void gather16B(const float* src, float* lds_dst) {
#ifdef HAVE_ASYNC_LDS
    __builtin_amdgcn_global_load_async_to_lds_b128(
        (__attribute__((address_space(1))) v4i*)(v4i*)(void*)src,
        (__attribute__((address_space(3))) v4i*)(v4i*)(void*)lds_dst,
        0, 0);
#else
    *(float4*)lds_dst = *(const float4*)src;
#endif
}

// Opaque zero for pinning the prologue gather's LDS address.
static __device__ __forceinline__ unsigned opaque_zero() {
    unsigned tok;
    asm volatile("s_mov_b32 %0, 0" : "=s"(tok));
    return tok;
}

// Pipeline sync point returning an opaque token that all of this epoch's LDS
// addresses must incorporate. Ordering is enforced by TRUE DATA DEPENDENCE:
//  - input `dep` (acc[0]) orders this asm after the previous epoch's WMMAs,
//    hence after the ds_loads feeding them (plus explicit s_wait_dscnt here)
//  - `tok` is defined only after the waits + barrier, so any LDS address
//    computed with it (gather dst, A-fragment reads) cannot be hoisted above
//    the barrier by any scheduler.
// Raw split-barrier asm (instead of __syncthreads) avoids the legalizer's
// conservative `s_wait_loadcnt 0x0` at every barrier: no cross-thread data
// flows through VGPR loads here, only LDS (dscnt) and async DMA (asynccnt).
static __device__ __forceinline__ unsigned sync_token(float dep) {
    unsigned tok;
#ifdef HAVE_ASYNC_LDS
    asm volatile("s_wait_dscnt 0x0\n\t"
                 "s_wait_asynccnt 0x0\n\t"
                 "s_barrier_signal -1\n\t"
                 "s_barrier_wait -1\n\t"
                 "s_mov_b32 %0, 0"
                 : "=s"(tok) : "v"(dep) : "memory");
#else
    asm volatile("s_wait_dscnt 0x0\n\t"
                 "s_barrier_signal -1\n\t"
                 "s_barrier_wait -1\n\t"
                 "s_mov_b32 %0, 0"
                 : "=s"(tok) : "v"(dep) : "memory");
#endif
    return tok;
}

// Prep: ws[0..31] = zero feature row (target for masked-out gathers);
// ws[32..] = weights repacked into per-lane-contiguous WMMA B-fragment order:
//   packed[(((k*4 + colt)*32 + lane)*16) + kk*2 + q] = W[k][4*kk + 2*(lane>>4) + q][16*colt + (lane&15)]
__global__ __launch_bounds__(128) void conv_prep(const float* __restrict__ w,
                                                 float* __restrict__ ws) {
    int i = blockIdx.x * 128 + threadIdx.x;
    if (i < 32) ws[i] = 0.0f;
    if (i < PACKED_FLOATS) {
        int q    = i & 1;
        int kk   = (i >> 1) & 7;
        int lane = (i >> 4) & 31;
        int colt = (i >> 9) & 3;
        int k    = i >> 11;
        int half = lane >> 4, lrow = lane & 15;
        ws[32 + i] = w[(size_t)k * (SPCIN * SPCOUT)
                       + (4 * kk + 2 * half + q) * SPCOUT
                       + 16 * colt + lrow];
    }
}

// Main kernel: one block = 16 output voxels, 4 waves; wave w -> cout [16w,16w+16).
// Double-buffered async LDS gather pipelined against the 8-step f32 WMMA chain;
// neighbor indices/masks prefetched two offsets ahead as raw values.
__global__ __launch_bounds__(128) void sparse_conv3d_wmma(
    const float* __restrict__ features,      // [n, 32]
    const int* __restrict__ nmap,            // [27, n]
    const unsigned char* __restrict__ nmask, // [27, n]
    const float* __restrict__ ws,            // zero row + packed weights
    float* __restrict__ out,                 // [n, 64]
    int n)
{
    __shared__ __align__(16) float sm[2][16 * ROWPAD];
    const float* zrow = ws;

    const int t    = threadIdx.x;
    const int tile = blockIdx.x;
    const int lane = t & 31;
    const int wv   = t >> 5;        // cout tile of this wave
    const int half = lane >> 4;
    const int lrow = lane & 15;

    // gather indexing: thread -> (row 0..15, 4-float cin chunk 0..7)
    const int gr    = t >> 3;
    const int gc    = t & 7;
    const int row_g = tile * 16 + gr;
    const bool rowok = (row_g < n);
    const int rg    = rowok ? row_g : (n - 1);   // clamped: all index/mask loads in-bounds

    // running pointers (advance by n per offset)
    const int*           nm_p = nmap + rg;
    const unsigned char* mk_p = nmask + rg;
    const float*         pwl  = ws + 32 + ((size_t)wv * 32 + lane) * 16;

    v8f acc = {0.f, 0.f, 0.f, 0.f, 0.f, 0.f, 0.f, 0.f};

    // ---- prologue: issue gather k=0; preload raw index/mask for k=1 ----
    {
        unsigned tok0 = opaque_zero();
        int nm0 = nm_p[0];
        int mr0 = mk_p[0];
        bool m  = rowok & (mr0 != 0);
        const float* src = (m ? features + (size_t)nm0 * SPCIN : zrow) + gc * 4;
        gather16B(src, &sm[0][gr * ROWPAD + gc * 4 + 4 * tok0]);
    }
    int nmN = nm_p[(size_t)n];
    int mrN = (int)mk_p[(size_t)n];
    nm_p += 2 * (size_t)n;
    mk_p += 2 * (size_t)n;

    int cur = 0;
    for (int k = 0; k < SPK; ++k) {
        // buf[cur] async writes complete + published; tok pins this epoch's
        // LDS traffic after the barrier via address data-dependence.
        // tok is scaled (x2 / x4) at use sites so alignment stays provable
        // and the DS accesses keep merging into b64/b128.
        const unsigned tok = sync_token(acc[0]);

        // issue gather k+1 (raw index/mask already resident; compare happens here)
        if (k + 1 < SPK) {
            bool m = rowok & (mrN != 0);
            const float* src = (m ? features + (size_t)nmN * SPCIN : zrow) + gc * 4;
            gather16B(src, &sm[cur ^ 1][gr * ROWPAD + gc * 4 + 4 * tok]);
        }

        // B fragments: 16 contiguous floats per lane -> 4x b128
        float breg[16];
        *(float4*)&breg[0]  = *(const float4*)(pwl + 0);
        *(float4*)&breg[4]  = *(const float4*)(pwl + 4);
        *(float4*)&breg[8]  = *(const float4*)(pwl + 8);
        *(float4*)&breg[12] = *(const float4*)(pwl + 12);
        pwl += 4 * 32 * 16;   // next offset's packed block

        // prefetch raw index/mask for k+2 AFTER the B loads (younger in LOADcnt;
        // with the raw split barrier these now stay in flight across it)
        if (k + 2 < SPK) {
            nmN = *nm_p;
            mrN = (int)*mk_p;
            nm_p += (size_t)n;
            mk_p += (size_t)n;
        }

        // WMMA: [16x32] (LDS) x [32x16] (regs) += [16x16]
        const float* smc = &sm[cur][lrow * ROWPAD + 2 * tok];
        #pragma unroll
        for (int kk = 0; kk < 8; ++kk) {
            v2f a, b;
            a.x = smc[kk * 4 + half * 2];
            a.y = smc[kk * 4 + half * 2 + 1];
            b.x = breg[2 * kk];
            b.y = breg[2 * kk + 1];
            acc = __builtin_amdgcn_wmma_f32_16x16x4_f32(
                false, a, false, b, (short)0, acc, false, false);
        }
        cur ^= 1;
    }

    // D: VGPR j -> row (j + 8*half), col lrow
    float* op = out + (size_t)tile * 16 * SPCOUT + wv * 16;
    #pragma unroll
    for (int j = 0; j < 8; ++j) {
        const int r = j + 8 * half;
        if (tile * 16 + r < n) op[(size_t)r * SPCOUT + lrow] = acc[j];
    }
}

// Fallback (no workspace): round-1 style, weights loaded strided, mask multiply.
__global__ __launch_bounds__(128) void sparse_conv3d_wmma_basic(
    const float* __restrict__ features, const int* __restrict__ nmap,
    const unsigned char* __restrict__ nmask, const float* __restrict__ weight,
    float* __restrict__ out, int n)
{
    __shared__ __align__(16) float sm[2][16 * ROWPAD];
    const int t = threadIdx.x, tile = blockIdx.x;
    const int lane = t & 31, wv = t >> 5, half = lane >> 4, lrow = lane & 15;
    const int gr = t >> 3, gc = t & 7;
    const int row_g = tile * 16 + gr;
    const bool rowok = (row_g < n);

    v8f acc = {0.f, 0.f, 0.f, 0.f, 0.f, 0.f, 0.f, 0.f};
    {
        float4 v = make_float4(0.f, 0.f, 0.f, 0.f);
        if (rowok) {
            int nm = nmap[row_g];
            float m = nmask[row_g] ? 1.0f : 0.0f;
            float4 f = *(const float4*)(features + (size_t)nm * SPCIN + gc * 4);
            v = make_float4(f.x * m, f.y * m, f.z * m, f.w * m);
        }
        *(float4*)&sm[0][gr * ROWPAD + gc * 4] = v;
    }
    int cur = 0;
    for (int k = 0; k < SPK; ++k) {
        __syncthreads();
        float4 nxt = make_float4(0.f, 0.f, 0.f, 0.f);
        const bool have = (k + 1 < SPK);
        if (have && rowok) {
            int nm = nmap[(size_t)(k + 1) * n + row_g];
            float m = nmask[(size_t)(k + 1) * n + row_g] ? 1.0f : 0.0f;
            float4 f = *(const float4*)(features + (size_t)nm * SPCIN + gc * 4);
            nxt = make_float4(f.x * m, f.y * m, f.z * m, f.w * m);
        }
        const float* Wk = weight + (size_t)k * (SPCIN * SPCOUT);
        const float* smc = &sm[cur][lrow * ROWPAD];
        #pragma unroll
        for (int kk = 0; kk < 8; ++kk) {
            const int kb = kk * 4 + half * 2;
            v2f a, b;
            a.x = smc[kb];
            a.y = smc[kb + 1];
            b.x = Wk[(size_t)kb * SPCOUT + wv * 16 + lrow];
            b.y = Wk[(size_t)(kb + 1) * SPCOUT + wv * 16 + lrow];
            acc = __builtin_amdgcn_wmma_f32_16x16x4_f32(
                false, a, false, b, (short)0, acc, false, false);
        }
        if (have) *(float4*)&sm[cur ^ 1][gr * ROWPAD + gc * 4] = nxt;
        cur ^= 1;
    }
    float* op = out + (size_t)tile * 16 * SPCOUT + wv * 16;
    #pragma unroll
    for (int j = 0; j < 8; ++j) {
        const int r = j + 8 * half;
        if (tile * 16 + r < n) op[(size_t)r * SPCOUT + lrow] = acc[j];
    }
}

extern "C" void kernel_launch(void* const* d_in, const int* in_sizes, int n_in,
                              void* d_out, int out_size, void* d_ws, size_t ws_size,
                              hipStream_t stream) {
    const float*         features = (const float*)d_in[0];
    const int*           nmap     = (const int*)d_in[1];
    const unsigned char* nmask    = (const unsigned char*)d_in[2];
    const float*         weight   = (const float*)d_in[3];
    float*               out      = (float*)d_out;

    const int n     = in_sizes[0] / SPCIN;   // 200000
    const int tiles = (n + 15) / 16;         // 12500

    if (ws_size >= WS_NEED) {
        conv_prep<<<dim3((PACKED_FLOATS + 127) / 128), dim3(128), 0, stream>>>(
            weight, (float*)d_ws);
        sparse_conv3d_wmma<<<dim3(tiles), dim3(128), 0, stream>>>(
            features, nmap, nmask, (const float*)d_ws, out, n);
    } else {
        sparse_conv3d_wmma_basic<<<dim3(tiles), dim3(128), 0, stream>>>(
            features, nmap, nmask, weight, out, n);
    }
}